// HungarianMatcher_5463198401024
// MI455X (gfx1250) — compile-verified
//
#include <hip/hip_runtime.h>
#include <cstdint>
#include <cstddef>

// Problem constants from the reference
#define NCC 20
#define NQC 27
#define BOXC 6
#define WAVES_PER_BLOCK 8   // 256 threads / wave32

#if __has_builtin(__builtin_amdgcn_global_load_async_to_lds_b32)
#define HAVE_ASYNC_LDS 1
#else
#define HAVE_ASYNC_LDS 0
#endif

// Address-space qualified pointer types for the async-to-LDS builtin
typedef __attribute__((address_space(1))) int g_int_t;   // global
typedef __attribute__((address_space(3))) int l_int_t;   // LDS

__global__ __launch_bounds__(256) void matcher_kernel(
    const float* __restrict__ logits,    // (bs, NC*NQ, 1)
    const float* __restrict__ boxes,     // (bs, NC*NQ, 6)
    const float* __restrict__ tgt,       // (bs, NC, 6)
    const int*   __restrict__ present,   // (bs, NC)
    const int*   __restrict__ ntq,       // scalar num_top_queries
    int*         __restrict__ matches,   // (bs, NC, NQ) int32
    float*       __restrict__ soft_out,  // (bs, NC, NQ) f32
    int total_pairs)                     // bs*NC
{
    __shared__ float s_tgt[WAVES_PER_BLOCK * BOXC];
    __shared__ int   s_pres[WAVES_PER_BLOCK];

    const int tid       = threadIdx.x;
    const int wave      = tid >> 5;
    const int lane      = tid & 31;
    const int base_pair = blockIdx.x * WAVES_PER_BLOCK;
    const int pair      = base_pair + wave;          // flattened (b*NC + c)
    const bool aq       = (lane < NQC);
    const bool inrange  = (pair < total_pairs);

    // ---- Prefetch this wave's pred rows early (overlaps the LDS staging) ----
    const size_t qidx = (size_t)pair * NQC + lane;
    if (inrange && aq) {
        __builtin_prefetch(boxes + qidx * BOXC, 0, 3);   // -> global_prefetch_b8
        __builtin_prefetch(logits + qidx, 0, 3);
    }

    // ---- Stage target boxes + presence into LDS via CDNA5 async-to-LDS ----
#if HAVE_ASYNC_LDS
    if (tid < WAVES_PER_BLOCK * BOXC) {                  // 48 dwords of tgt boxes
        const int p = base_pair + tid / BOXC;
        if (p < total_pairs) {
            const float* g = tgt + (size_t)base_pair * BOXC + tid;
            __builtin_amdgcn_global_load_async_to_lds_b32(
                (g_int_t*)g,
                (l_int_t*)(&s_tgt[tid]),
                0, 0);
        }
    } else if (tid >= 64 && tid < 64 + WAVES_PER_BLOCK) { // 8 dwords of presence
        const int p = base_pair + (tid - 64);
        if (p < total_pairs) {
            const int* g = present + p;
            __builtin_amdgcn_global_load_async_to_lds_b32(
                (g_int_t*)g,
                (l_int_t*)(&s_pres[tid - 64]),
                0, 0);
        }
    }
#if __has_builtin(__builtin_amdgcn_s_wait_asynccnt)
    __builtin_amdgcn_s_wait_asynccnt(0);
#else
    asm volatile("s_wait_asynccnt 0" ::: "memory");
#endif
#else
    if (tid < WAVES_PER_BLOCK * BOXC) {
        const int p = base_pair + tid / BOXC;
        if (p < total_pairs) s_tgt[tid] = tgt[(size_t)base_pair * BOXC + tid];
    } else if (tid >= 64 && tid < 64 + WAVES_PER_BLOCK) {
        const int p = base_pair + (tid - 64);
        if (p < total_pairs) s_pres[tid - 64] = present[p];
    }
#endif
    __syncthreads();

    if (!inrange) return;

    // ---- Per-lane cost evaluation (lane == query index) ----
    float t[BOXC];
#pragma unroll
    for (int i = 0; i < BOXC; ++i) t[i] = s_tgt[wave * BOXC + i];
    const int pres = s_pres[wave];

    // Each 24 B box row is 8-byte aligned (648*pair + 24*lane), so load as
    // 3x float2 -> global_load_b64 (half the VMEM ops of scalar loads).
    float bx[BOXC] = {0.f, 0.f, 0.f, 0.f, 0.f, 0.f};
    float lg = 0.f;
    if (aq) {
        const float2* bp2 = reinterpret_cast<const float2*>(boxes + qidx * BOXC);
#pragma unroll
        for (int j = 0; j < 3; ++j) {
            const float2 v = bp2[j];
            bx[2 * j]     = v.x;
            bx[2 * j + 1] = v.y;
        }
        lg = logits[qidx];
    }

    // cost_bbox: L1 on raw cxcyczwhd
    float l1 = 0.f;
#pragma unroll
    for (int i = 0; i < BOXC; ++i) l1 += fabsf(bx[i] - t[i]);

    // cost_class: -sigmoid(logit)  (precise expf: feeds an argmin)
    const float cls = -1.0f / (1.0f + expf(-lg));

    // GIoU3D on clip(pred, 0) vs tgt, both converted cxcyczwhd -> xyzxyz
    float cb[BOXC];
#pragma unroll
    for (int i = 0; i < BOXC; ++i) cb[i] = fmaxf(bx[i], 0.f);

    const float volp = cb[3] * cb[4] * cb[5];
    const float volt = t[3] * t[4] * t[5];
    float inter = 1.f, hull = 1.f;
#pragma unroll
    for (int i = 0; i < 3; ++i) {
        const float pmin = cb[i] - 0.5f * cb[i + 3];
        const float pmax = cb[i] + 0.5f * cb[i + 3];
        const float tmin = t[i]  - 0.5f * t[i + 3];
        const float tmax = t[i]  + 0.5f * t[i + 3];
        inter *= fmaxf(fminf(pmax, tmax) - fmaxf(pmin, tmin), 0.f);
        hull  *= fmaxf(fmaxf(pmax, tmax) - fminf(pmin, tmin), 0.f);
    }
    const float uni  = volp + volt - inter;
    const float giou = inter / uni - (hull - uni) / hull;
    const float cg   = -giou;                      // cost_giou
    const float C    = 5.f * l1 + 2.f * cls + 2.f * cg;

    // ---- Wave32 cross-lane reductions (inactive lanes -> sentinels) ----
    const float INF = __builtin_inff();
    float cmax = aq ? cg : -INF;
    float cmin = aq ? cg :  INF;
    float myC  = aq ? C  :  INF;
#pragma unroll
    for (int off = 16; off > 0; off >>= 1) {
        cmax = fmaxf(cmax, __shfl_xor(cmax, off, 32));
        cmin = fminf(cmin, __shfl_xor(cmin, off, 32));
    }

    // top-k via iterative stable argmin (reference tie-break: lowest index)
    int k = ntq[0];
    if (k < 0) k = 0;
    if (k > NQC) k = NQC;
    bool picked = false;
    for (int it = 0; it < k; ++it) {
        float v = myC;
        int   ix = lane;
#pragma unroll
        for (int off = 16; off > 0; off >>= 1) {
            const float ov = __shfl_xor(v, off, 32);
            const int   oi = __shfl_xor(ix, off, 32);
            if (ov < v || (ov == v && oi < ix)) { v = ov; ix = oi; }
        }
        if (lane == ix) { picked = true; myC = INF; }
    }

    // ---- Write outputs (coalesced across the 27 active lanes) ----
    if (aq) {
        matches[qidx] = pres ? (picked ? 1 : 0) : 0;
        const float sf = fmaxf((cg - cmax) / (cmin - cmax), 0.f);
        soft_out[qidx] = pres ? sf : -1.0f;
    }
}

extern "C" void kernel_launch(void* const* d_in, const int* in_sizes, int n_in,
                              void* d_out, int out_size, void* d_ws, size_t ws_size,
                              hipStream_t stream) {
    (void)n_in; (void)d_ws; (void)ws_size; (void)out_size;
    const float* logits  = (const float*)d_in[0];  // (bs, NC*NQ, 1) f32
    const float* boxes   = (const float*)d_in[1];  // (bs, NC*NQ, 6) f32
    const float* tgtb    = (const float*)d_in[2];  // (bs, NC, 6)    f32
    const int*   pres    = (const int*)d_in[3];    // (bs, NC)       i32
    const int*   ntq     = (const int*)d_in[4];    // scalar

    const int bs          = in_sizes[0] / (NCC * NQC);
    const int total_pairs = bs * NCC;
    const int n_elems     = total_pairs * NQC;

    int*   matches = (int*)d_out;                  // first output (int32 bits)
    float* soft    = (float*)d_out + n_elems;      // second output (f32)

    const int blocks = (total_pairs + WAVES_PER_BLOCK - 1) / WAVES_PER_BLOCK;
    matcher_kernel<<<blocks, 256, 0, stream>>>(logits, boxes, tgtb, pres, ntq,
                                               matches, soft, total_pairs);
}